// DSGCNBlock_10823317586641
// MI455X (gfx1250) — compile-verified
//
#include <hip/hip_runtime.h>
#include <math.h>

typedef float v2f __attribute__((ext_vector_type(2)));
typedef float v8f __attribute__((ext_vector_type(8)));
typedef __attribute__((address_space(3))) float lds_f;

#define B_    32
#define T_    128
#define N_    47
#define CIN   192
#define COUT  256
#define KK    3
#define CC    512               // combined GEMM cols: 256 h + 256 residual
#define EPSF  1e-5f

// workspace layout (float offsets)
#define OFF_SCALE ((size_t)0)
#define OFF_WCAT  ((size_t)256)
#define OFF_H     (OFF_WCAT + (size_t)N_*CC*CIN)                 // (B,N,COUT,T)
#define OFF_RES   (OFF_H    + (size_t)B_*N_*COUT*T_)             // (B,T,N,COUT)
#define OFF_GN    (OFF_RES  + (size_t)B_*T_*N_*COUT)             // (B,T,N,COUT)

// ---------------------------------------------------------------------------
// 1) scale[k,n] = rowsum(A_eff) with abs-rowsum normalization clamped at 1
// ---------------------------------------------------------------------------
__global__ void k_scale(const float* __restrict__ A, const float* __restrict__ adj,
                        float* __restrict__ scale) {
  int i = threadIdx.x;
  if (i >= KK * N_) return;
  int k = i / N_, n = i % N_;
  const float* ar = A   + ((size_t)k * N_ + n) * N_;
  const float* rr = adj + ((size_t)k * N_ + n) * N_;
  float s = 0.f, as = 0.f;
  for (int m = 0; m < N_; ++m) {
    float v = ar[m] + tanhf(rr[m]) * 0.3f;
    s += v; as += fabsf(v);
  }
  scale[i] = s / fmaxf(as, 1.0f);
}

// ---------------------------------------------------------------------------
// 2) Wcat[n][o][c]: o<256 -> sum_k scale[k,n]*dw[k,c]*W_pw[o,k*C+c]
//                   o>=256 -> W_res[o-256][c]
// ---------------------------------------------------------------------------
__global__ void k_wcat(const float* __restrict__ scale, const float* __restrict__ dw,
                       const float* __restrict__ Wpw, const float* __restrict__ Wres,
                       float* __restrict__ wcat) {
  size_t idx = (size_t)blockIdx.x * blockDim.x + threadIdx.x;
  if (idx >= (size_t)N_ * CC * CIN) return;
  int n   = (int)(idx / ((size_t)CC * CIN));
  int rem = (int)(idx % ((size_t)CC * CIN));
  int o = rem / CIN, c = rem % CIN;
  float w;
  if (o < COUT) {
    w = 0.f;
#pragma unroll
    for (int k = 0; k < KK; ++k)
      w += scale[k * N_ + n] * dw[k * CIN + c] * Wpw[(size_t)o * (KK * CIN) + k * CIN + c];
  } else {
    w = Wres[(size_t)(o - COUT) * CIN + c];
  }
  wcat[idx] = w;
}

// ---------------------------------------------------------------------------
// 3) fused GEMM: per node n, [4096 x 192] @ [192 x 512] using f32 WMMA 16x16x4
//    Per-wave tile 32M x 64N: each B fragment feeds 2 WMMAs.
//    waves 0..3 -> h cols (stored (B,N,C,T)); waves 4..7 -> residual cols.
//    x tile staged to LDS with gfx1250 async-to-LDS DMA (ASYNCcnt).
// ---------------------------------------------------------------------------
__global__ __launch_bounds__(256) void k_gemm(const float* __restrict__ x,
                                              const float* __restrict__ wcat,
                                              float* __restrict__ h_ws,
                                              float* __restrict__ res_ws) {
  __shared__ float xs[32 * CIN];                    // 24 KB x tile (32 rows)
  const int n     = blockIdx.y;
  const int mbase = blockIdx.x * 32;                // row tile (b,t) flat
  const int b     = mbase >> 7;                     // / T_
  const int tbase = mbase & (T_ - 1);
  const int tid   = threadIdx.x;

  // async-copy x tile (32 rows x 192 f32 = 1536 float4) straight into LDS.
  // LDS offset derived from the real pointer via AS(3) cast: gives the
  // wave-relative byte offset AND makes xs escape so the compiler keeps
  // the subsequent ds_load A-fragment reads alive.
#pragma unroll
  for (int i = 0; i < 6; ++i) {
    int j  = i * 256 + tid;                         // float4 index [0,1536)
    int r  = j / (CIN / 4);
    int c4 = j % (CIN / 4);
    unsigned long long gaddr = (unsigned long long)(const void*)(
        x + (size_t)((b * T_ + tbase + r) * N_ + n) * CIN + c4 * 4);
    unsigned lds_off = (unsigned)(unsigned long long)(lds_f*)(xs + r * CIN + c4 * 4);
    asm volatile("global_load_async_to_lds_b128 %0, %1, off"
                 :: "v"(lds_off), "v"(gaddr)
                 : "memory");
  }
  asm volatile("s_wait_asynccnt 0x0" ::: "memory");
  __syncthreads();

  const int lane  = tid & 31;
  const int wave  = tid >> 5;
  const int l15   = lane & 15;
  const int hi2   = (lane >= 16) ? 2 : 0;           // K-half select (f32 16x16x4 layout)
  const int obase = wave * 64;
  const float* wn = wcat + (size_t)n * CC * CIN;

  v8f acc[2][4];
#pragma unroll
  for (int m = 0; m < 2; ++m)
#pragma unroll
    for (int i = 0; i < 4; ++i)
      acc[m][i] = (v8f){0.f,0.f,0.f,0.f,0.f,0.f,0.f,0.f};

  for (int kk = 0; kk < CIN; kk += 4) {
    if (kk + 16 < CIN)
      __builtin_prefetch(wn + (size_t)(obase + l15) * CIN + kk + 16, 0, 3);
    // A fragments for the two 16-row halves of the 32-row tile
    v2f av0 = *reinterpret_cast<const v2f*>(xs + l15 * CIN + kk + hi2);
    v2f av1 = *reinterpret_cast<const v2f*>(xs + (16 + l15) * CIN + kk + hi2);
#pragma unroll
    for (int t4 = 0; t4 < 4; ++t4) {
      int col = obase + t4 * 16 + l15;              // B col N
      v2f bv = *reinterpret_cast<const v2f*>(wn + (size_t)col * CIN + kk + hi2);
      acc[0][t4] = __builtin_amdgcn_wmma_f32_16x16x4_f32(
          false, av0, false, bv, (short)0, acc[0][t4], false, false);
      acc[1][t4] = __builtin_amdgcn_wmma_f32_16x16x4_f32(
          false, av1, false, bv, (short)0, acc[1][t4], false, false);
    }
  }

  const int thi = (lane >= 16) ? 8 : 0;             // C/D layout: vgpr r -> M=r(+8)
  if (wave < 4) {
    // lane owns fixed o, 8 consecutive t -> contiguous 32B stores
#pragma unroll
    for (int m = 0; m < 2; ++m)
#pragma unroll
      for (int t4 = 0; t4 < 4; ++t4) {
        int col = obase + t4 * 16 + l15;
        float* dst = h_ws + ((size_t)(b * N_ + n) * COUT + col) * T_
                   + tbase + m * 16 + thi;
        float4 f0 = {acc[m][t4][0], acc[m][t4][1], acc[m][t4][2], acc[m][t4][3]};
        float4 f1 = {acc[m][t4][4], acc[m][t4][5], acc[m][t4][6], acc[m][t4][7]};
        reinterpret_cast<float4*>(dst)[0] = f0;
        reinterpret_cast<float4*>(dst)[1] = f1;
      }
  } else {
    // residual straight into (B,T,N,COUT)
#pragma unroll
    for (int m = 0; m < 2; ++m)
#pragma unroll
      for (int t4 = 0; t4 < 4; ++t4) {
        int o = obase + t4 * 16 + l15 - COUT;
#pragma unroll
        for (int r = 0; r < 8; ++r) {
          int t = tbase + m * 16 + r + thi;
          res_ws[((size_t)(b * T_ + t) * N_ + n) * COUT + o] = acc[m][t4][r];
        }
      }
  }
}

// ---------------------------------------------------------------------------
// 4) depthwise temporal conv (KT=3, pad 1) + GroupNorm.
//    thread = channel; group = 32 channels = exactly one wave32 -> shfl stats
// ---------------------------------------------------------------------------
__global__ __launch_bounds__(256) void k_convgn(const float* __restrict__ h_ws,
                                                const float* __restrict__ conv_w,
                                                const float* __restrict__ gn_gamma,
                                                const float* __restrict__ gn_beta,
                                                float* __restrict__ gn_ws) {
  const int bn = blockIdx.x;
  const int b = bn / N_, n = bn % N_;
  const int o = threadIdx.x;
  const float* src = h_ws + ((size_t)bn * COUT + o) * T_;

  float v[T_];
#pragma unroll
  for (int t4 = 0; t4 < T_ / 4; ++t4) {
    float4 f = reinterpret_cast<const float4*>(src)[t4];
    v[t4 * 4 + 0] = f.x; v[t4 * 4 + 1] = f.y;
    v[t4 * 4 + 2] = f.z; v[t4 * 4 + 3] = f.w;
  }
  const float w0 = conv_w[o * 3 + 0], w1 = conv_w[o * 3 + 1], w2 = conv_w[o * 3 + 2];
  float sum = 0.f, sq = 0.f, prev = 0.f;
#pragma unroll
  for (int t = 0; t < T_; ++t) {
    float cur = v[t];
    float nxt = (t < T_ - 1) ? v[t + 1] : 0.f;
    float c = w0 * prev + w1 * cur + w2 * nxt;
    prev = cur;
    v[t] = c;
    sum += c; sq += c * c;
  }
  // group stats across one wave (32 channels x 128 t)
#pragma unroll
  for (int off = 16; off > 0; off >>= 1) {
    sum += __shfl_xor(sum, off, 32);
    sq  += __shfl_xor(sq,  off, 32);
  }
  const float inv = 1.0f / (32.0f * T_);
  float mu  = sum * inv;
  float var = sq * inv - mu * mu;
  float rs  = rsqrtf(var + EPSF);
  float ga = gn_gamma[o], be = gn_beta[o];
#pragma unroll
  for (int t = 0; t < T_; ++t) {
    float yv = (v[t] - mu) * rs * ga + be;
    gn_ws[((size_t)(b * T_ + t) * N_ + n) * COUT + o] = yv;   // coalesced over o
  }
}

// ---------------------------------------------------------------------------
// 5) y = gn + residual; LayerNorm(C) ; exact GELU
// ---------------------------------------------------------------------------
__global__ __launch_bounds__(256) void k_lngelu(const float* __restrict__ gn_ws,
                                                const float* __restrict__ res_ws,
                                                const float* __restrict__ ln_gamma,
                                                const float* __restrict__ ln_beta,
                                                float* __restrict__ out) {
  const size_t row = blockIdx.x;
  const int o = threadIdx.x;
  float y = gn_ws[row * COUT + o] + res_ws[row * COUT + o];
  float s = y, q = y * y;
#pragma unroll
  for (int off = 16; off > 0; off >>= 1) {
    s += __shfl_xor(s, off, 32);
    q += __shfl_xor(q, off, 32);
  }
  __shared__ float ss[8], sqv[8];
  int wave = o >> 5, lane = o & 31;
  if (lane == 0) { ss[wave] = s; sqv[wave] = q; }
  __syncthreads();
  float ts = 0.f, tq = 0.f;
#pragma unroll
  for (int w = 0; w < 8; ++w) { ts += ss[w]; tq += sqv[w]; }
  float mu  = ts / (float)COUT;
  float var = tq / (float)COUT - mu * mu;
  float rs  = rsqrtf(var + EPSF);
  float z = (y - mu) * rs * ln_gamma[o] + ln_beta[o];
  out[row * COUT + o] = 0.5f * z * (1.0f + erff(z * 0.70710678118654752f));
}

// ---------------------------------------------------------------------------
extern "C" void kernel_launch(void* const* d_in, const int* in_sizes, int n_in,
                              void* d_out, int out_size, void* d_ws, size_t ws_size,
                              hipStream_t stream) {
  const float* x    = (const float*)d_in[0];
  const float* A    = (const float*)d_in[1];
  const float* dw   = (const float*)d_in[2];
  const float* adj  = (const float*)d_in[3];
  const float* Wpw  = (const float*)d_in[4];
  const float* cw   = (const float*)d_in[5];
  const float* gng  = (const float*)d_in[6];
  const float* gnb  = (const float*)d_in[7];
  const float* lng  = (const float*)d_in[8];
  const float* lnb  = (const float*)d_in[9];
  const float* Wres = (const float*)d_in[10];
  float* ws  = (float*)d_ws;
  float* out = (float*)d_out;

  k_scale<<<1, 160, 0, stream>>>(A, adj, ws + OFF_SCALE);

  size_t wtot = (size_t)N_ * CC * CIN;
  k_wcat<<<(unsigned)((wtot + 255) / 256), 256, 0, stream>>>(
      ws + OFF_SCALE, dw, Wpw, Wres, ws + OFF_WCAT);

  dim3 gg((B_ * T_) / 32, N_);
  k_gemm<<<gg, 256, 0, stream>>>(x, ws + OFF_WCAT, ws + OFF_H, ws + OFF_RES);

  k_convgn<<<B_ * N_, COUT, 0, stream>>>(ws + OFF_H, cw, gng, gnb, ws + OFF_GN);

  k_lngelu<<<B_ * T_ * N_, COUT, 0, stream>>>(ws + OFF_GN, ws + OFF_RES, lng, lnb, out);
}